// SAFB_25623774888343
// MI455X (gfx1250) — compile-verified
//
#include <hip/hip_runtime.h>
#include <hip/hip_bf16.h>

typedef float v2f __attribute__((ext_vector_type(2)));
typedef float v8f __attribute__((ext_vector_type(8)));

#define CC   256
#define HH   128
#define WW   128
#define HWSZ (HH * WW)

__device__ __forceinline__ int reflect128(int i) {
    if (i < 0) i = -i;
    if (i > 127) i = 254 - i;
    return i;
}
__device__ __forceinline__ float sigmoidf_(float v) {
    return 1.0f / (1.0f + __expf(-v));
}

// ---------------------------------------------------------------------------
// Kernel A: per-(b,c) plane reductions over initial = x + y.
// 128 threads; thread t owns column w=t (coalesced loads), rows via LDS atomics.
// ---------------------------------------------------------------------------
__global__ void safb_reduce(const float* __restrict__ x, const float* __restrict__ y,
                            float* __restrict__ rowsum, float* __restrict__ colsum,
                            float* __restrict__ gapsum) {
    __shared__ float rows[HH];
    __shared__ float gtot;
    const int t  = threadIdx.x;          // 0..127
    const int bc = blockIdx.x;           // b*256 + c
    const size_t base = (size_t)bc * HWSZ;

    rows[t] = 0.0f;
    if (t == 0) gtot = 0.0f;
    __syncthreads();

    float colAcc = 0.0f;
    for (int i = 0; i < HH; ++i) {
        size_t gi = base + (size_t)i * WW + t;
        float v = x[gi] + y[gi];
        colAcc += v;
        float s = v;
        #pragma unroll
        for (int off = 16; off > 0; off >>= 1) s += __shfl_xor(s, off, 32);
        if ((t & 31) == 0) atomicAdd(&rows[i], s);
    }
    __syncthreads();

    rowsum[(size_t)bc * HH + t] = rows[t];
    colsum[(size_t)bc * WW + t] = colAcc;

    float s = rows[t];
    #pragma unroll
    for (int off = 16; off > 0; off >>= 1) s += __shfl_xor(s, off, 32);
    if ((t & 31) == 0) atomicAdd(&gtot, s);
    __syncthreads();
    if (t == 0) gapsum[bc] = gtot;
}

// ---------------------------------------------------------------------------
// Kernel B: channel attention MLP: gap -> relu(W1 g + b1) -> W2 h + b2.
// One block of 256 threads.
// ---------------------------------------------------------------------------
__global__ void safb_cattn(const float* __restrict__ gapsum,
                           const float* __restrict__ w1, const float* __restrict__ b1,
                           const float* __restrict__ w2, const float* __restrict__ b2,
                           float* __restrict__ cattn) {
    __shared__ float hid[8 * 32];
    const int t = threadIdx.x;  // 0..255
    {
        const int b = t >> 5, j = t & 31;
        float acc = b1[j];
        const float* g = gapsum + b * CC;
        const float* w = w1 + j * CC;
        for (int c = 0; c < CC; ++c)
            acc = fmaf(g[c] * (1.0f / 16384.0f), w[c], acc);
        hid[b * 32 + j] = fmaxf(acc, 0.0f);
    }
    __syncthreads();
    const int o = t;
    for (int b = 0; b < 8; ++b) {
        float acc = b2[o];
        const float* h = &hid[b * 32];
        const float* w = w2 + o * 32;
        #pragma unroll
        for (int j = 0; j < 32; ++j) acc = fmaf(h[j], w[j], acc);
        cattn[b * CC + o] = acc;
    }
}

// ---------------------------------------------------------------------------
// Kernel C: fused spatial attn + depthwise 7x7 (reflect) + double sigmoid +
// blend:  result = initial + y + p2*(x - y)
// Tile 32x32 with 3-halo (38x38) in LDS; halo recomputed as x+y.
// dwconv tap:  (w1*sh*sw + w0) * initial,  + cattn * sum(w1) + pa_b
// ---------------------------------------------------------------------------
#define TS 32
#define HS 38
#define HP 40

__global__ void safb_fuse(const float* __restrict__ x, const float* __restrict__ y,
                          const float* __restrict__ rowsum, const float* __restrict__ colsum,
                          const float* __restrict__ cattn,
                          const float* __restrict__ pa_w, const float* __restrict__ pa_b,
                          float* __restrict__ result) {
    __shared__ float sInit[HS * HP];
    __shared__ float sSS[HS * HP];
    __shared__ float sh[HS], sw[HS];
    __shared__ float wk0[49], wk1[49];
    __shared__ float w1sum;

    const int t   = threadIdx.x;        // 0..255
    const int blk = blockIdx.x;
    const int tw  = blk & 3;
    const int th  = (blk >> 2) & 3;
    const int bc  = blk >> 4;           // b*256 + c
    const int c   = bc & (CC - 1);
    const size_t pbase = (size_t)bc * HWSZ;
    const int h0 = th * TS - 3;
    const int w0 = tw * TS - 3;

    if (t < 98) {
        float wv = pa_w[c * 98 + t];
        if (t < 49) wk0[t] = wv; else wk1[t - 49] = wv;
    }
    if (t >= 128 && t < 128 + HS) {
        int gh = reflect128(h0 + (t - 128));
        sh[t - 128] = sigmoidf_(rowsum[(size_t)bc * HH + gh] * (1.0f / 128.0f));
    }
    if (t >= 192 && t < 192 + HS) {
        int gw = reflect128(w0 + (t - 192));
        sw[t - 192] = sigmoidf_(colsum[(size_t)bc * WW + gw] * (1.0f / 128.0f));
    }
    for (int e = t; e < HS * HS; e += 256) {
        int lh = e / HS, lw = e - lh * HS;
        int gh = reflect128(h0 + lh), gw = reflect128(w0 + lw);
        size_t gi = pbase + (size_t)gh * WW + gw;
        sInit[lh * HP + lw] = x[gi] + y[gi];
    }
    __syncthreads();

    if (t == 0) {
        float s = 0.0f;
        #pragma unroll
        for (int k = 0; k < 49; ++k) s += wk1[k];
        w1sum = s;
    }
    for (int e = t; e < HS * HS; e += 256) {
        int lh = e / HS, lw = e - lh * HS;
        sSS[lh * HP + lw] = sh[lh] * sw[lw];
    }
    __syncthreads();

    const float bias = fmaf(cattn[bc], w1sum, pa_b[c]);
    for (int p = 0; p < 4; ++p) {
        int pi = t + p * 256;              // 0..1023
        int i = pi >> 5, j = pi & 31;      // output pixel in tile
        float acc = bias;
        #pragma unroll
        for (int kh = 0; kh < 7; ++kh) {
            #pragma unroll
            for (int kw = 0; kw < 7; ++kw) {
                int li = (i + kh) * HP + (j + kw);
                float tv = fmaf(wk1[kh * 7 + kw], sSS[li], wk0[kh * 7 + kw]);
                acc = fmaf(tv, sInit[li], acc);
            }
        }
        float p2 = sigmoidf_(sigmoidf_(acc));
        size_t gi = pbase + (size_t)(th * TS + i) * WW + (tw * TS + j);
        float xx = x[gi], yy = y[gi];
        float init = sInit[(i + 3) * HP + (j + 3)];
        result[gi] = init + yy + p2 * (xx - yy);
    }
}

// ---------------------------------------------------------------------------
// Kernel D: 1x1 conv as WMMA f32 GEMM.
// Per batch b:  out[o, n] = sum_c W[o,c] * R[c, n] + bias[o],  n over HW.
// One wave computes a 16(O) x 64(N) tile = 4 accumulators, K loop step 4.
// A layout (16x4): lane = 16*kg + m, vgpr v -> K = 2*kg + v
// B layout (4x16): vgpr v, lanes0-15 K=v (N=m), lanes16-31 K=v+2
// D layout (16x16): vgpr v -> row M = v + 8*kg, col N = m
// ---------------------------------------------------------------------------
__global__ void safb_gemm(const float* __restrict__ R, const float* __restrict__ Wm,
                          const float* __restrict__ bias, float* __restrict__ out) {
    const int wid  = blockIdx.x * 8 + (threadIdx.x >> 5);
    const int lane = threadIdx.x & 31;
    const int o16  = wid & 15;
    const int n64  = (wid >> 4) & 255;
    const int b    = wid >> 12;
    const int m    = lane & 15;
    const int kg   = lane >> 4;       // 0 or 1
    const int obase = o16 * 16;
    const int nbase = n64 * 64;
    const float* Rb = R + (size_t)b * CC * HWSZ;
    float* Ob = out + (size_t)b * CC * HWSZ;

    v8f acc0, acc1, acc2, acc3;
    #pragma unroll
    for (int v = 0; v < 8; ++v) {
        float bv = bias[obase + v + 8 * kg];
        acc0[v] = bv; acc1[v] = bv; acc2[v] = bv; acc3[v] = bv;
    }

    const float* wrow = Wm + (size_t)(obase + m) * CC;   // row-major [256,256]
    for (int k0 = 0; k0 < CC; k0 += 4) {
        const int kk = k0 + 2 * kg;
        v2f a;
        a.x = wrow[kk];
        a.y = wrow[kk + 1];
        const float* rp = Rb + (size_t)kk * HWSZ + nbase + m;
        v2f b0, b1, b2, b3;
        b0.x = rp[0];   b0.y = rp[HWSZ];
        b1.x = rp[16];  b1.y = rp[HWSZ + 16];
        b2.x = rp[32];  b2.y = rp[HWSZ + 32];
        b3.x = rp[48];  b3.y = rp[HWSZ + 48];
        acc0 = __builtin_amdgcn_wmma_f32_16x16x4_f32(false, a, false, b0, (short)0, acc0, false, false);
        acc1 = __builtin_amdgcn_wmma_f32_16x16x4_f32(false, a, false, b1, (short)0, acc1, false, false);
        acc2 = __builtin_amdgcn_wmma_f32_16x16x4_f32(false, a, false, b2, (short)0, acc2, false, false);
        acc3 = __builtin_amdgcn_wmma_f32_16x16x4_f32(false, a, false, b3, (short)0, acc3, false, false);
    }

    #pragma unroll
    for (int v = 0; v < 8; ++v) {
        int o = obase + v + 8 * kg;
        float* op = Ob + (size_t)o * HWSZ + nbase + m;
        op[0]  = acc0[v];
        op[16] = acc1[v];
        op[32] = acc2[v];
        op[48] = acc3[v];
    }
}

// ---------------------------------------------------------------------------
extern "C" void kernel_launch(void* const* d_in, const int* in_sizes, int n_in,
                              void* d_out, int out_size, void* d_ws, size_t ws_size,
                              hipStream_t stream) {
    const float* x      = (const float*)d_in[0];
    const float* y      = (const float*)d_in[1];
    const float* ca_w1  = (const float*)d_in[2];
    const float* ca_b1  = (const float*)d_in[3];
    const float* ca_w2  = (const float*)d_in[4];
    const float* ca_b2  = (const float*)d_in[5];
    const float* pa_w   = (const float*)d_in[6];
    const float* pa_b   = (const float*)d_in[7];
    const float* conv_w = (const float*)d_in[8];
    const float* conv_b = (const float*)d_in[9];
    float* out = (float*)d_out;

    float* ws      = (float*)d_ws;
    float* result  = ws;                       // 8*256*128*128 = 33,554,432 f32
    float* rowsum  = result + 33554432;        // 8*256*128 = 262,144 f32
    float* colsum  = rowsum + 262144;          // 262,144 f32
    float* gapsum  = colsum + 262144;          // 2,048 f32
    float* cattn   = gapsum + 2048;            // 2,048 f32

    // A: reductions (one block per (b,c) plane)
    safb_reduce<<<8 * CC, 128, 0, stream>>>(x, y, rowsum, colsum, gapsum);
    // B: channel attention MLP
    safb_cattn<<<1, 256, 0, stream>>>(gapsum, ca_w1, ca_b1, ca_w2, ca_b2, cattn);
    // C: fused spatial/pixel attention + blend -> result
    safb_fuse<<<8 * CC * 16, 256, 0, stream>>>(x, y, rowsum, colsum, cattn, pa_w, pa_b, result);
    // D: 1x1 conv GEMM via WMMA f32 (8 waves per block, 32768 wave-tiles)
    safb_gemm<<<4096, 256, 0, stream>>>(result, conv_w, conv_b, out);
}